// SeqTagger_34694745817302
// MI455X (gfx1250) — compile-verified
//
#include <hip/hip_runtime.h>
#include <hip/hip_bf16.h>
#include <cstdint>

// Problem constants (match reference)
#define KTAGS 128
#define TSEQ  512
#define BBAT  512
#define TS    130          // padded LDS row stride (floats) for transposed transition
#define BOS_TAG 0
#define EOS_TAG 1
#define NEGV  (-10000.0f)

static __device__ __forceinline__ unsigned lds_addr_of(const void* p) {
    // Generic pointer to a __shared__ object: low 32 bits are the LDS byte offset
    // (LDS aperture lives entirely in ADDR[63:32] on CDNA5).
    return (unsigned)(uintptr_t)p;
}

__global__ __launch_bounds__(KTAGS)
void viterbi_kernel(const float* __restrict__ X,          // [B,T,K]
                    const float* __restrict__ trans,      // [K,K]
                    float* __restrict__ out,              // [B] scores ++ [B,T] paths (as float)
                    unsigned char* __restrict__ bpw)      // [B,T,K] backpointers (u8)
{
    __shared__ __align__(16) float s_transT[KTAGS * TS];  // transT[n*TS+p] = trans[p][n]
    __shared__ __align__(16) float s_fv[2][KTAGS];
    __shared__ __align__(16) float s_emit[2][KTAGS];
    __shared__ float s_rv[KTAGS];
    __shared__ int   s_ri[KTAGS];

    const int tid = threadIdx.x;      // next-tag owned by this thread
    const int b   = blockIdx.x;

    const unsigned emitBase0 = lds_addr_of(&s_emit[0][0]);
    const unsigned emitBase1 = lds_addr_of(&s_emit[1][0]);

    // ---- Prologue: kick off async load of emission row t=0 (wave 0 only) ----
    if (tid < 32) {
        unsigned ldsa = emitBase0 + (unsigned)tid * 16u;
        unsigned voff = ((unsigned)b * TSEQ + 0u) * (KTAGS * 4u) + (unsigned)tid * 16u;
        asm volatile("global_load_async_to_lds_b128 %0, %1, %2"
                     :
                     : "v"(ldsa), "v"(voff), "s"(X)
                     : "memory");
    }

    // ---- Transpose transition into LDS (coalesced global reads, conflict-free LDS writes) ----
    for (int p = 0; p < KTAGS; ++p) {
        s_transT[tid * TS + p] = trans[p * KTAGS + tid];
    }

    // ---- init viterbi vars ----
    s_fv[0][tid] = (tid == BOS_TAG) ? 0.0f : NEGV;

    if (tid < 32) {
        asm volatile("s_wait_asynccnt 0" ::: "memory");
    }
    __syncthreads();

    // ---- Forward pass ----
    for (int t = 0; t < TSEQ; ++t) {
        const int cur = t & 1;
        const int nxt = cur ^ 1;

        // Prefetch emission row for t+1 into the other LDS buffer (async DMA, wave 0).
        if ((t + 1 < TSEQ) && tid < 32) {
            unsigned ldsa = (nxt ? emitBase1 : emitBase0) + (unsigned)tid * 16u;
            unsigned voff = ((unsigned)(b * TSEQ + t + 1)) * (KTAGS * 4u) + (unsigned)tid * 16u;
            asm volatile("global_load_async_to_lds_b128 %0, %1, %2"
                         :
                         : "v"(ldsa), "v"(voff), "s"(X)
                         : "memory");
        }

        const float* fv = s_fv[cur];
        const float* tr = &s_transT[tid * TS];

        float best = -3.0e38f;
        int   bp   = 0;
        #pragma unroll
        for (int p = 0; p < KTAGS; p += 4) {
            const float4 f  = *(const float4*)(fv + p);      // broadcast read
            const float2 a0 = *(const float2*)(tr + p);      // bank-conflict-free b64
            const float2 a2 = *(const float2*)(tr + p + 2);
            float v;
            v = f.x + a0.x; if (v > best) { best = v; bp = p;     }
            v = f.y + a0.y; if (v > best) { best = v; bp = p + 1; }
            v = f.z + a2.x; if (v > best) { best = v; bp = p + 2; }
            v = f.w + a2.y; if (v > best) { best = v; bp = p + 3; }
        }

        // coalesced u8 backpointer store
        bpw[((size_t)b * TSEQ + t) * KTAGS + tid] = (unsigned char)bp;

        // emission added after the max (as in reference)
        s_fv[nxt][tid] = best + s_emit[cur][tid];

        if (tid < 32) {
            asm volatile("s_wait_asynccnt 0" ::: "memory");  // prefetch landed
        }
        __syncthreads();
    }

    // ---- Terminal scores: fv + trans[:, EOS];  transT[EOS][n] == trans[n][EOS] ----
    const float* fvF = s_fv[TSEQ & 1];
    float term = fvF[tid] + s_transT[EOS_TAG * TS + tid];
    s_rv[tid] = term;
    s_ri[tid] = tid;
    __syncthreads();

    // argmax reduction, ties -> lower index (matches jnp.argmax first-occurrence)
    for (int s = KTAGS / 2; s > 0; s >>= 1) {
        if (tid < s) {
            float v2 = s_rv[tid + s];
            if (v2 > s_rv[tid]) { s_rv[tid] = v2; s_ri[tid] = s_ri[tid + s]; }
        }
        __syncthreads();
    }

    // ---- Backtrace (sequential per block; 512 independent chains across the grid) ----
    if (tid == 0) {
        out[b] = s_rv[0];                       // path_scores[b]
        float* paths = out + BBAT;              // best_paths flattened [B,T]
        int tag = s_ri[0];
        paths[(size_t)b * TSEQ + (TSEQ - 1)] = (float)tag;
        const unsigned char* bp = bpw;
        for (int t = TSEQ - 1; t >= 1; --t) {
            tag = bp[((size_t)b * TSEQ + t) * KTAGS + tag];
            paths[(size_t)b * TSEQ + (t - 1)] = (float)tag;
        }
    }
}

extern "C" void kernel_launch(void* const* d_in, const int* in_sizes, int n_in,
                              void* d_out, int out_size, void* d_ws, size_t ws_size,
                              hipStream_t stream) {
    const float* X     = (const float*)d_in[0];   // [512,512,128] f32
    const float* trans = (const float*)d_in[1];   // [128,128] f32 (already has BOS/EOS masking)
    float* out = (float*)d_out;                   // [512] scores ++ [512*512] paths
    unsigned char* bpw = (unsigned char*)d_ws;    // 32 MB backpointer scratch

    viterbi_kernel<<<BBAT, KTAGS, 0, stream>>>(X, trans, out, bpw);
}